// FNOModel3d_4501125726907
// MI455X (gfx1250) — compile-verified
//
#include <hip/hip_runtime.h>
#include <hip/hip_bf16.h>
#include <math.h>

// FNO3d on gfx1250: truncated separable DFTs as V_WMMA_F32_16X16X4_F32 GEMMs,
// LDS staging via Tensor Data Mover (hardware row padding) + async global->LDS.

typedef __attribute__((ext_vector_type(2))) float v2f;
typedef __attribute__((ext_vector_type(8))) float v8f;
typedef __attribute__((ext_vector_type(4))) unsigned int u32x4;
typedef __attribute__((ext_vector_type(8))) int i32x8;
typedef __attribute__((ext_vector_type(4))) int i32x4;

#define ZERO8 {0.f,0.f,0.f,0.f,0.f,0.f,0.f,0.f}
#define AS1 __attribute__((address_space(1)))
#define AS3 __attribute__((address_space(3)))

#if __has_builtin(__builtin_amdgcn_tensor_load_to_lds) && \
    __has_builtin(__builtin_amdgcn_s_wait_tensorcnt)
#define HAVE_TDM 1
#else
#define HAVE_TDM 0
#endif

#if __has_builtin(__builtin_amdgcn_global_load_async_to_lds_b32) && \
    __has_builtin(__builtin_amdgcn_s_wait_asynccnt)
#define HAVE_ASYNC 1
#else
#define HAVE_ASYNC 0
#endif

#if HAVE_ASYNC
// Param 0 is a non-const addrspace(1) int* (per hipcc diagnostic); route both
// pointers through integer casts to get the right addrspace/type/constness.
__device__ __forceinline__ void async_ld_b32(const float* g, const float* lds_elem) {
  __builtin_amdgcn_global_load_async_to_lds_b32(
      (AS1 int*)(unsigned long long)(uintptr_t)g,
      (AS3 int*)(unsigned)(uintptr_t)lds_elem, 0, 0);
}
#endif

#if HAVE_TDM
// One TDM 2D tile load: global fp32 tile (dim0 x dim1 elements, row stride
// stride0 elements) -> LDS at byte offset lds_byte. If pad_iv >= 0, HW inserts
// +1 dword after every (2 << pad_iv) dwords stored (pad_iv=5 -> every 64 dw,
// giving 65-dword LDS rows; pad_iv=3 -> every 16 dw, 17-dword rows).
// D# bitfields per CDNA5 ISA ch.8 (group0 128b, group1 256b).
__device__ __forceinline__ void tdm_load_f32_2d(unsigned lds_byte,
                                                const float* gptr,
                                                unsigned dim0, unsigned dim1,
                                                unsigned stride0, int pad_iv) {
  unsigned long long ga = (unsigned long long)(uintptr_t)gptr;
  u32x4 g0;
  g0[0] = 1u;                                            // count=1, user D#
  g0[1] = lds_byte;                                      // lds_addr (bytes)
  g0[2] = (unsigned)(ga & 0xFFFFFFFFull);                // global_addr[31:0]
  g0[3] = (unsigned)((ga >> 32) & 0x1FFFFFFull) | (2u << 30); // addr[56:32]|type=2
  i32x8 g1;
  unsigned e0 = (2u << 16);                              // data_size=4B
  if (pad_iv >= 0) e0 |= (1u << 20) | ((unsigned)pad_iv << 22); // pad_amount=0 (+1dw)
  g1[0] = (int)e0;
  g1[1] = (int)(dim0 << 16);                             // tensor_dim0[15:0]
  g1[2] = (int)((dim0 >> 16) | (dim1 << 16));            // dim0 hi | tensor_dim1 lo
  g1[3] = (int)((dim1 >> 16) | (dim0 << 16));            // dim1 hi | tile_dim0
  g1[4] = (int)dim1;                                     // tile_dim1 (tile_dim2=0)
  g1[5] = (int)stride0;                                  // tensor_dim0_stride lo32
  g1[6] = 0;
  g1[7] = 0;
  i32x4 gz; gz[0] = 0; gz[1] = 0; gz[2] = 0; gz[3] = 0;
#if __clang_major__ >= 23
  i32x8 gz8; gz8[0]=0; gz8[1]=0; gz8[2]=0; gz8[3]=0; gz8[4]=0; gz8[5]=0; gz8[6]=0; gz8[7]=0;
  __builtin_amdgcn_tensor_load_to_lds(g0, g1, gz, gz, gz8, 0);
#else
  __builtin_amdgcn_tensor_load_to_lds(g0, g1, gz, gz, 0);
#endif
}
#endif

// D(16x16) += A(16xK) * B(Kx16); A,B row-major in LDS with strides lda/ldb.
__device__ __forceinline__ v8f wmma_acc_f32(const float* __restrict__ As, int lda,
                                            const float* __restrict__ Bs, int ldb,
                                            int K, v8f acc) {
  const int lane = threadIdx.x & 31;
  const int m  = lane & 15;
  const int kh = (lane >> 4) << 1; // 0 or 2
  for (int k = 0; k < K; k += 4) {
    v2f a, b;
    a.x = As[m * lda + k + kh + 0];
    a.y = As[m * lda + k + kh + 1];
    b.x = Bs[(k + kh + 0) * ldb + m];
    b.y = Bs[(k + kh + 1) * ldb + m];
    acc = __builtin_amdgcn_wmma_f32_16x16x4_f32(false, a, false, b, (short)0, acc,
                                                false, false);
  }
  return acc;
}

__device__ __forceinline__ void wmma_store(float* __restrict__ D, int ldd, v8f acc) {
  const int lane = threadIdx.x & 31;
  const int n  = lane & 15;
  const int rb = (lane >> 4) << 3; // 0 or 8
#pragma unroll
  for (int r = 0; r < 8; ++r) D[(rb + r) * ldd + n] = acc[r];
}

// ---------------- basis tables (24KB at ws[0..6143]) -------------------------
__global__ void k_tables(float* __restrict__ ws) {
  const int t = threadIdx.x; // 1024 threads
  const float PI2 = 6.283185307179586f;
  { int z = t >> 4, c = t & 15, k = c & 7;
    float th = PI2 * (float)(k * z) / 64.0f;
    ws[t] = (c < 8) ? cosf(th) : -sinf(th); }
  { int kidx = t >> 6, y = t & 63; float f = (float)(kidx - 8);
    float th = PI2 * f * (float)y / 64.0f;
    ws[1024 + t] = cosf(th);
    ws[2048 + t] = -sinf(th); }
  { int x = t >> 4, c = t & 15; float f = (float)(c - 8);
    float th = PI2 * f * (float)x / 64.0f;
    ws[3072 + t] = cosf(th) * (1.0f / 64.0f);
    ws[4096 + t] = sinf(th) * (1.0f / 64.0f); }
  { int c = t >> 6, z = t & 63; int k = c & 7;
    float wk = ((k == 0) ? 1.0f : 2.0f) / 64.0f;
    float th = PI2 * (float)(k * z) / 64.0f;
    ws[5120 + t] = (c < 8) ? wk * cosf(th) : -wk * sinf(th); }
}

// ---------------- conv_in (1->32) + zero-pad to 64^3 -------------------------
__global__ void k_conv_in(const float* __restrict__ X, const float* __restrict__ w,
                          const float* __restrict__ bias, float* __restrict__ H) {
  long t = (long)blockIdx.x * blockDim.x + threadIdx.x;
  if (t >= (long)2 * 32 * 262144) return;
  int z = (int)(t & 63); long r = t >> 6;
  int y = (int)(r & 63); r >>= 6;
  int x = (int)(r & 63); r >>= 6;
  int c = (int)(r & 31); int b = (int)(r >> 5);
  float v = 0.f;
  if (x >= 8 && x < 56 && y >= 8 && y < 56 && z >= 8 && z < 56) {
    long xi = (((long)b * 48 + (x - 8)) * 48 + (y - 8)) * 48 + (z - 8);
    v = X[xi] * w[c] + bias[c];
  }
  H[t] = v;
}

// ---------------- forward truncated DFT along z: 64 -> 8 complex -------------
// Per (b,c,x) plane: D(64y x 16) = H(64y x 64z) * FZb(64z x 16).
__global__ void k_fwdZ(const float* __restrict__ H, const float* __restrict__ tabs,
                       float* __restrict__ Sz) {
  __shared__ float As[64 * 65];
  __shared__ float Bs[64 * 16];
  const int bcx = blockIdx.x;                 // (b*32+c)*64 + x
  const long hbase = (long)bcx * 4096;
  for (int i = threadIdx.x; i < 1024; i += blockDim.x) Bs[i] = tabs[i];
#if HAVE_TDM
  if ((threadIdx.x >> 5) == 0) {   // wave 0 issues the DMA; TDM ignores EXEC
    tdm_load_f32_2d((unsigned)(uintptr_t)(void*)As, H + hbase, 64u, 64u, 64u, 5);
    __builtin_amdgcn_s_wait_tensorcnt(0);
  }
#else
  for (int i = threadIdx.x; i < 4096; i += blockDim.x)
    As[(i >> 6) * 65 + (i & 63)] = H[hbase + i];
#endif
  __syncthreads();
  const int wv = threadIdx.x >> 5;
  v8f acc = ZERO8;
  acc = wmma_acc_f32(As + wv * 16 * 65, 65, Bs, 16, 64, acc);
  wmma_store(Sz + (long)bcx * 1024 + wv * 256, 16, acc);
}

// ---------------- forward DFT along y: 64 -> 16 complex ----------------------
__global__ void k_fwdY(const float* __restrict__ Sz, const float* __restrict__ tabs,
                       float* __restrict__ Sy) {
  __shared__ float Are[16 * 65], Aim[16 * 65], Bs[64 * 16], E[2][16 * 17];
  const int bcx = blockIdx.x;
  for (int i = threadIdx.x; i < 1024; i += blockDim.x) {
    int r = i >> 6, c = i & 63;
    Are[r * 65 + c] = tabs[1024 + i];
    Aim[r * 65 + c] = tabs[2048 + i];
    Bs[i] = Sz[(long)bcx * 1024 + i];
  }
  __syncthreads();
  const int wv = threadIdx.x >> 5; // 0: Re-basis, 1: Im-basis
  v8f acc = ZERO8;
  acc = wmma_acc_f32(wv ? Aim : Are, 65, Bs, 16, 64, acc);
  wmma_store(&E[wv][0], 17, acc);
  __syncthreads();
  float* out = Sy + (long)bcx * 256;
  for (int i = threadIdx.x; i < 256; i += blockDim.x) {
    int ky = i >> 4, c = i & 15;
    float v = (c < 8) ? (E[0][ky * 17 + c] - E[1][ky * 17 + c + 8])
                      : (E[0][ky * 17 + c] + E[1][ky * 17 + c - 8]);
    out[i] = v;
  }
}

// ---------------- forward DFT along x: 64 -> 16 complex ----------------------
__global__ void k_fwdX(const float* __restrict__ Sy, const float* __restrict__ tabs,
                       float* __restrict__ S) {
  __shared__ float Are[16 * 65], Aim[16 * 65], Bs[64 * 16], E[2][16 * 17];
  const int blk = blockIdx.x;      // (b*32+c)*16 + ky
  const int ky = blk & 15, bc = blk >> 4;
  for (int i = threadIdx.x; i < 1024; i += blockDim.x) {
    int r = i >> 6, c = i & 63;
    Are[r * 65 + c] = tabs[1024 + i];
    Aim[r * 65 + c] = tabs[2048 + i];
    int x = i >> 4, cc = i & 15;
    Bs[i] = Sy[((long)bc * 64 + x) * 256 + ky * 16 + cc];
  }
  __syncthreads();
  const int wv = threadIdx.x >> 5;
  v8f acc = ZERO8;
  acc = wmma_acc_f32(wv ? Aim : Are, 65, Bs, 16, 64, acc);
  wmma_store(&E[wv][0], 17, acc);
  __syncthreads();
  for (int i = threadIdx.x; i < 256; i += blockDim.x) {
    int kx = i >> 4, c = i & 15;
    float v = (c < 8) ? (E[0][kx * 17 + c] - E[1][kx * 17 + c + 8])
                      : (E[0][kx * 17 + c] + E[1][kx * 17 + c - 8]);
    S[(long)bc * 4096 + kx * 256 + ky * 16 + c] = v;
  }
}

// ---------------- per-mode complex 32x32 channel mix -------------------------
__global__ void k_spectral(const float* __restrict__ S, const float* __restrict__ W,
                           float* __restrict__ S2) {
  int t = blockIdx.x * blockDim.x + threadIdx.x;
  if (t >= 2 * 32 * 16 * 16 * 8) return;
  int kz = t & 7; int r = t >> 3;
  int ky = r & 15; r >>= 4;
  int kx = r & 15; r >>= 4;
  int o = r & 31; int b = r >> 5;
  const long sbase = (long)b * 32 * 4096 + kx * 256 + ky * 16 + kz;
  const long wbase = (long)o * 4096 + kx * 256 + (long)ky * 16 + kz * 2;
  float oRe = 0.f, oIm = 0.f;
  for (int i = 0; i < 32; ++i) {
    const float* sp = S + sbase + (long)i * 4096;
    const float* wp = W + wbase + (long)i * 131072;
    __builtin_prefetch(wp + 131072, 0, 1); // global_prefetch_b8
    float sRe = sp[0], sIm = sp[8];
    float wRe = wp[0], wIm = wp[1];
    oRe += sRe * wRe - sIm * wIm;
    oIm += sRe * wIm + sIm * wRe;
  }
  long obase = ((long)b * 32 + o) * 4096 + kx * 256 + ky * 16;
  S2[obase + kz] = oRe;
  S2[obase + kz + 8] = oIm;
}

// ---------------- inverse DFT along x: 16 -> 64 ------------------------------
__global__ void k_invX(const float* __restrict__ S2, const float* __restrict__ tabs,
                       float* __restrict__ G1) {
  __shared__ float Are[64 * 17], Aim[64 * 17], Bs[16 * 16];
  __shared__ float E1[64 * 17], E2[64 * 17];
  const int blk = blockIdx.x;  // (b*32+o)*16 + ky
  const int ky = blk & 15, bo = blk >> 4;
  for (int i = threadIdx.x; i < 1024; i += blockDim.x) {
    int r = i >> 4, c = i & 15;
    Are[r * 17 + c] = tabs[3072 + i];
    Aim[r * 17 + c] = tabs[4096 + i];
  }
  for (int i = threadIdx.x; i < 256; i += blockDim.x) {
    int kx = i >> 4, c = i & 15;
    Bs[i] = S2[(long)bo * 4096 + kx * 256 + ky * 16 + c];
  }
  __syncthreads();
  const int wv = threadIdx.x >> 5; // 4 waves: one 16-row x-tile each
  v8f a1 = ZERO8, a2 = ZERO8;
  a1 = wmma_acc_f32(Are + wv * 16 * 17, 17, Bs, 16, 16, a1);
  a2 = wmma_acc_f32(Aim + wv * 16 * 17, 17, Bs, 16, 16, a2);
  wmma_store(E1 + wv * 16 * 17, 17, a1);
  wmma_store(E2 + wv * 16 * 17, 17, a2);
  __syncthreads();
  for (int i = threadIdx.x; i < 1024; i += blockDim.x) {
    int x = i >> 4, c = i & 15;
    float v = (c < 8) ? (E1[x * 17 + c] - E2[x * 17 + c + 8])
                      : (E1[x * 17 + c] + E2[x * 17 + c - 8]);
    G1[((long)bo * 64 + x) * 256 + ky * 16 + c] = v;
  }
}

// ---------------- inverse DFT along y: 16 -> 64 ------------------------------
__global__ void k_invY(const float* __restrict__ G1, const float* __restrict__ tabs,
                       float* __restrict__ G2) {
  __shared__ float Are[64 * 17], Aim[64 * 17], Bs[16 * 16];
  __shared__ float E1[64 * 17], E2[64 * 17];
  const int blk = blockIdx.x;  // (b*32+o)*64 + x
  for (int i = threadIdx.x; i < 1024; i += blockDim.x) {
    int r = i >> 4, c = i & 15;
    Are[r * 17 + c] = tabs[3072 + i];
    Aim[r * 17 + c] = tabs[4096 + i];
  }
  for (int i = threadIdx.x; i < 256; i += blockDim.x)
    Bs[i] = G1[(long)blk * 256 + i];   // rows ky, cols 16
  __syncthreads();
  const int wv = threadIdx.x >> 5;
  v8f a1 = ZERO8, a2 = ZERO8;
  a1 = wmma_acc_f32(Are + wv * 16 * 17, 17, Bs, 16, 16, a1);
  a2 = wmma_acc_f32(Aim + wv * 16 * 17, 17, Bs, 16, 16, a2);
  wmma_store(E1 + wv * 16 * 17, 17, a1);
  wmma_store(E2 + wv * 16 * 17, 17, a2);
  __syncthreads();
  for (int i = threadIdx.x; i < 1024; i += blockDim.x) {
    int y = i >> 4, c = i & 15;
    float v = (c < 8) ? (E1[y * 17 + c] - E2[y * 17 + c + 8])
                      : (E1[y * 17 + c] + E2[y * 17 + c - 8]);
    G2[(long)blk * 1024 + i] = v;
  }
}

// ---------------- pointwise 32x32 conv as WMMA GEMM --------------------------
__global__ void k_conv32(const float* __restrict__ H, const float* __restrict__ Wc,
                         float* __restrict__ Hn) {
  __shared__ float As[32 * 33], Bs[32 * 33];
  const int blk = blockIdx.x;
  const int b = blk >> 13;
  const long s0 = (long)(blk & 8191) * 32;
#if HAVE_ASYNC
  for (int i = threadIdx.x; i < 1024; i += blockDim.x) {
    int r = i >> 5, c = i & 31;
    As[r * 33 + c] = Wc[i];
    async_ld_b32(H + ((long)b * 32 + r) * 262144 + s0 + c, &Bs[r * 33 + c]);
  }
  __builtin_amdgcn_s_wait_asynccnt(0);
#else
  for (int i = threadIdx.x; i < 1024; i += blockDim.x) {
    int r = i >> 5, c = i & 31;
    As[r * 33 + c] = Wc[i];
    Bs[r * 33 + c] = H[((long)b * 32 + r) * 262144 + s0 + c];
  }
#endif
  __syncthreads();
  const int wv = threadIdx.x >> 5;
  const int ot = wv & 1, st = wv >> 1;
  v8f acc = ZERO8;
  acc = wmma_acc_f32(As + ot * 16 * 33, 33, Bs + st * 16, 33, 32, acc);
  const int lane = threadIdx.x & 31;
  const int n = lane & 15, rb = (lane >> 4) << 3;
#pragma unroll
  for (int r = 0; r < 8; ++r) {
    int o = ot * 16 + rb + r;
    Hn[((long)b * 32 + o) * 262144 + s0 + st * 16 + n] = acc[r];
  }
}

// ---------------- inverse C2R along z (16 -> 64) fused with += ---------------
__global__ void k_invZ(const float* __restrict__ G2, const float* __restrict__ tabs,
                       float* __restrict__ Hn) {
  __shared__ float As[64 * 17], Bz[16 * 65];
  const int blk = blockIdx.x;  // (b*32+o)*64 + x
#if HAVE_TDM
  if ((threadIdx.x >> 5) == 0) {
    tdm_load_f32_2d((unsigned)(uintptr_t)(void*)As, G2 + (long)blk * 1024,
                    16u, 64u, 16u, 3);          // 16-dw rows -> 17-dw LDS rows
    tdm_load_f32_2d((unsigned)(uintptr_t)(void*)Bz, tabs + 5120,
                    64u, 16u, 64u, 5);          // 64-dw rows -> 65-dw LDS rows
    __builtin_amdgcn_s_wait_tensorcnt(0);
  }
#else
  for (int i = threadIdx.x; i < 1024; i += blockDim.x) {
    As[(i >> 4) * 17 + (i & 15)] = G2[(long)blk * 1024 + i];
    Bz[(i >> 6) * 65 + (i & 63)] = tabs[5120 + i];
  }
#endif
  __syncthreads();
  const int wv = threadIdx.x >> 5;  // M-tile
  const int lane = threadIdx.x & 31;
  const int n = lane & 15, rb = (lane >> 4) << 3;
  const long base = (long)blk * 4096;
  for (int nt = 0; nt < 4; ++nt) {
    v8f acc = ZERO8;
    acc = wmma_acc_f32(As + wv * 16 * 17, 17, Bz + nt * 16, 65, 16, acc);
#pragma unroll
    for (int r = 0; r < 8; ++r) {
      long off = base + (long)(wv * 16 + rb + r) * 64 + nt * 16 + n;
      Hn[off] += acc[r];
    }
  }
}

// ---------------- batchnorm ---------------------------------------------------
__global__ void k_bn_zero(float* __restrict__ stats) {
  if (threadIdx.x < 64) stats[threadIdx.x] = 0.f;
}

__global__ void k_bn_stats(const float* __restrict__ H, float* __restrict__ stats) {
  __shared__ float red[2][256];
  const int blk = blockIdx.x;
  const int bc = blk >> 6;
  const long base = (long)bc * 262144 + (long)(blk & 63) * 4096;
  float s = 0.f, s2 = 0.f;
  for (int i = threadIdx.x; i < 4096; i += 256) {
    float v = H[base + i];
    s += v; s2 += v * v;
  }
  red[0][threadIdx.x] = s; red[1][threadIdx.x] = s2;
  __syncthreads();
  for (int w = 128; w > 0; w >>= 1) {
    if (threadIdx.x < w) {
      red[0][threadIdx.x] += red[0][threadIdx.x + w];
      red[1][threadIdx.x] += red[1][threadIdx.x + w];
    }
    __syncthreads();
  }
  if (threadIdx.x == 0) {
    int c = bc & 31;
    atomicAdd(&stats[c], red[0][0]);
    atomicAdd(&stats[32 + c], red[1][0]);
  }
}

__global__ void k_bn_apply(float* __restrict__ H, const float* __restrict__ stats,
                           const float* __restrict__ gamma,
                           const float* __restrict__ beta) {
  long t = (long)blockIdx.x * blockDim.x + threadIdx.x;
  if (t >= (long)2 * 32 * 262144) return;
  int c = (int)((t >> 18) & 31);
  const float invN = 1.0f / 524288.0f;
  float mean = stats[c] * invN;
  float var = stats[32 + c] * invN - mean * mean;
  float v = H[t];
  float xn = (v - mean) * rsqrtf(var + 1e-5f);
  float y = xn * gamma[c] + beta[c];
  H[t] = 0.5f * y * (1.0f + erff(y * 0.70710678118654752f));
}

// ---------------- crop + conv_out (32 -> 1) ----------------------------------
__global__ void k_conv_out(const float* __restrict__ H, const float* __restrict__ Wo,
                           const float* __restrict__ Bo, float* __restrict__ out) {
  int t = blockIdx.x * blockDim.x + threadIdx.x;
  if (t >= 2 * 48 * 48 * 48) return;
  int k = t % 48; int r = t / 48;
  int j = r % 48; r /= 48;
  int i = r % 48; int b = r / 48;
  long base = (long)b * 32 * 262144 + (long)(i + 8) * 4096 + (j + 8) * 64 + (k + 8);
  float acc = Bo[0];
  for (int c = 0; c < 32; ++c) acc += H[base + (long)c * 262144] * Wo[c];
  out[t] = acc;
}

// ---------------- host launcher ----------------------------------------------
extern "C" void kernel_launch(void* const* d_in, const int* in_sizes, int n_in,
                              void* d_out, int out_size, void* d_ws, size_t ws_size,
                              hipStream_t stream) {
  const float* X    = (const float*)d_in[0];
  const float* Wci  = (const float*)d_in[1];
  const float* Bci  = (const float*)d_in[2];
  const float* Wfft = (const float*)d_in[3];
  const float* Wcv  = (const float*)d_in[4];
  const float* Gam  = (const float*)d_in[5];
  const float* Bet  = (const float*)d_in[6];
  const float* Wco  = (const float*)d_in[7];
  const float* Bco  = (const float*)d_in[8];
  float* out = (float*)d_out;
  float* ws = (float*)d_ws;

  float* tabs  = ws;                        // 6144
  float* stats = ws + 6144;                 // 64
  float* hA    = ws + 8192;                 // 16,777,216 (64MB)
  float* hB    = hA + 16777216;             // 16,777,216 (64MB)
  float* Sz    = hB + 16777216;             // 4,194,304  (16MB)
  float* Sy    = Sz + 4194304;              // 1,048,576  (4MB)
  float* Sb    = Sy + 1048576;              // 262,144    (1MB)
  float* S2    = Sb + 262144;               // 262,144    (1MB)
  float* G1    = Sy;                        // reuse (Sy dead after fwdX)
  float* G2    = Sz;                        // reuse (Sz dead after fwdY)

  k_tables<<<1, 1024, 0, stream>>>(tabs);
  k_conv_in<<<65536, 256, 0, stream>>>(X, Wci, Bci, hA);

  for (int blk = 0; blk < 4; ++blk) {
    k_fwdZ<<<4096, 128, 0, stream>>>(hA, tabs, Sz);
    k_fwdY<<<4096, 64, 0, stream>>>(Sz, tabs, Sy);
    k_fwdX<<<1024, 64, 0, stream>>>(Sy, tabs, Sb);
    k_spectral<<<512, 256, 0, stream>>>(Sb, Wfft + (long)blk * 4194304, S2);
    k_invX<<<1024, 128, 0, stream>>>(S2, tabs, G1);
    k_invY<<<4096, 128, 0, stream>>>(G1, tabs, G2);
    k_conv32<<<16384, 128, 0, stream>>>(hA, Wcv + blk * 1024, hB);
    k_invZ<<<4096, 128, 0, stream>>>(G2, tabs, hB);
    k_bn_zero<<<1, 64, 0, stream>>>(stats);
    k_bn_stats<<<4096, 256, 0, stream>>>(hB, stats);
    k_bn_apply<<<65536, 256, 0, stream>>>(hB, stats, Gam + blk * 32, Bet + blk * 32);
    float* tmp = hA; hA = hB; hB = tmp;
  }
  k_conv_out<<<864, 256, 0, stream>>>(hA, Wco, Bco, out);
}